// DiscreteKeyValueBottleneck_3745211482262
// MI455X (gfx1250) — compile-verified
//
#include <hip/hip_runtime.h>

typedef float v2f __attribute__((ext_vector_type(2)));
typedef float v8f __attribute__((ext_vector_type(8)));

#define BB 16
#define CC 8
#define TT 256
#define DD 64
#define KK 4096
#define VV 64

// ---------------------------------------------------------------------------
// Kernel 1: bias[c][k] = 0.5 * ||keys[c][k]||^2   (argmin(d2) == argmax(xk - bias))
// ---------------------------------------------------------------------------
__global__ void k2_kernel(const float* __restrict__ keys, float* __restrict__ k2w) {
    int gid = blockIdx.x * blockDim.x + threadIdx.x;  // one key per thread
    if (gid >= CC * KK) return;
    const float4* p = (const float4*)(keys + (size_t)gid * DD);
    float s = 0.f;
#pragma unroll
    for (int i = 0; i < DD / 4; ++i) {
        float4 q = p[i];
        s += q.x * q.x + q.y * q.y + q.z * q.z + q.w * q.w;
    }
    k2w[gid] = 0.5f * s;
}

// ---------------------------------------------------------------------------
// Kernel 2: fused f32-WMMA GEMM + running argmax (== argmin of distance).
// Grid: (16 M-tiles of 256 rows, 8 codebooks). Block: 256 threads = 8 waves.
// Each wave owns TWO 16-row A tiles (64 VGPRs, loaded once, reused for all
// 256 key tiles) so every LDS B-fragment feeds 2 WMMAs. Key tiles are
// double-buffered in LDS (stride-68 pad -> conflict-free ds_load_b64) with a
// 2-deep register prefetch pipeline and a single barrier per iteration;
// global_prefetch warms L1 8 tiles ahead.
// ---------------------------------------------------------------------------
__global__ __launch_bounds__(256)
void argmax_kernel(const float* __restrict__ batch, const float* __restrict__ keys,
                   const float* __restrict__ k2w, int* __restrict__ idxw) {
    const int c     = blockIdx.y;       // codebook
    const int mtile = blockIdx.x;       // 256-row tile of the 4096 (b,t) rows
    const int tid   = threadIdx.x;
    const int wave  = tid >> 5;
    const int lane  = tid & 31;
    const int lr    = lane & 15;        // row (A) / col (B,C) within 16
    const int lh    = lane >> 4;        // lane half selects K-pair

    __shared__ float tile[2][16 * 68];  // double-buffered 16x64 key tile, pad 68

    const float* kbase = keys + (size_t)c * KK * DD;
    const float* bbase = k2w + (size_t)c * KK;

    // ---- preload A fragments for two 16-row subtiles (ISA 16x4 f32 A layout) ----
    v2f a0[16], a1[16];
    {
        int m0 = mtile * 256 + wave * 32 + lr;
        int m1 = m0 + 16;
        const float* x0 = batch + (((size_t)(m0 >> 8) * CC + c) * TT + (m0 & (TT - 1))) * DD;
        const float* x1 = batch + (((size_t)(m1 >> 8) * CC + c) * TT + (m1 & (TT - 1))) * DD;
#pragma unroll
        for (int s = 0; s < 16; ++s) {
            a0[s] = *(const v2f*)(x0 + 4 * s + 2 * lh);
            a1[s] = *(const v2f*)(x1 + 4 * s + 2 * lh);
        }
    }

    float best0[8], best1[8];
    int   bidx0[8], bidx1[8];
#pragma unroll
    for (int r = 0; r < 8; ++r) {
        best0[r] = -3.4e38f; bidx0[r] = 0;
        best1[r] = -3.4e38f; bidx1[r] = 0;
    }

    // staging slot owned by this thread: one float4 of the 16x64 tile
    const int srow = tid >> 4;            // 0..15 (key row within tile)
    const int scol = (tid & 15) * 4;      // 0..60
    const float* gsrc = kbase + (size_t)srow * DD + scol;

    constexpr int NT = KK / 16;

    // prologue: stage tile 0, prefetch tile 1 into registers
    {
        float4 q0 = *(const float4*)(gsrc);
        float* d = &tile[0][srow * 68 + scol];
        d[0] = q0.x; d[1] = q0.y; d[2] = q0.z; d[3] = q0.w;
    }
    float  bcur = bbase[lr];
    float  bnext = 0.f;
    float4 qn = *(const float4*)(gsrc + 16 * DD);
    __syncthreads();

    for (int nt = 0; nt < NT; ++nt) {
        const int cur = nt & 1;

        // store tile nt+1 into the alternate buffer (its readers finished at
        // iteration nt-1; the barrier at the end of nt-1 covers them)
        if (nt + 1 < NT) {
            float* d = &tile[1 - cur][srow * 68 + scol];
            d[0] = qn.x; d[1] = qn.y; d[2] = qn.z; d[3] = qn.w;
            bnext = bbase[(nt + 1) * 16 + lr];
        }
        // refill the register pipeline 2 tiles ahead; warm L1 8 tiles ahead
        if (nt + 2 < NT) qn = *(const float4*)(gsrc + (size_t)(nt + 2) * 16 * DD);
        if (nt + 8 < NT) __builtin_prefetch(gsrc + (size_t)(nt + 8) * 16 * DD, 0, 1);

        // 16 K-steps; each LDS B fragment feeds two accumulators
        v8f acc0 = {}, acc1 = {};
#pragma unroll
        for (int s = 0; s < 16; ++s) {
            v2f bf = *(const v2f*)(&tile[cur][lr * 68 + 4 * s + 2 * lh]);
            acc0 = __builtin_amdgcn_wmma_f32_16x16x4_f32(
                false, a0[s], false, bf, (short)0, acc0, false, false);
            acc1 = __builtin_amdgcn_wmma_f32_16x16x4_f32(
                false, a1[s], false, bf, (short)0, acc1, false, false);
        }

        // running argmax of (xk - 0.5*k2); tie -> lower index (matches argmin)
        const int id = nt * 16 + lr;
#pragma unroll
        for (int r = 0; r < 8; ++r) {
            float v0 = acc0[r] - bcur;
            if (v0 > best0[r] || (v0 == best0[r] && id < bidx0[r])) {
                best0[r] = v0; bidx0[r] = id;
            }
            float v1 = acc1[r] - bcur;
            if (v1 > best1[r] || (v1 == best1[r] && id < bidx1[r])) {
                best1[r] = v1; bidx1[r] = id;
            }
        }
        bcur = bnext;
        __syncthreads();
    }

    // reduce the 16 column-candidates per row (shuffles stay within lane halves)
#pragma unroll
    for (int sub = 0; sub < 2; ++sub) {
        float* best = sub ? best1 : best0;
        int*   bidx = sub ? bidx1 : bidx0;
#pragma unroll
        for (int r = 0; r < 8; ++r) {
            float v = best[r];
            int   id = bidx[r];
#pragma unroll
            for (int off = 8; off >= 1; off >>= 1) {
                float ov = __shfl_xor(v, off, 32);
                int   oid = __shfl_xor(id, off, 32);
                if (ov > v || (ov == v && oid < id)) { v = ov; id = oid; }
            }
            best[r] = v; bidx[r] = id;
        }
        if (lr == 0) {  // lane 0 -> rows +0..7, lane 16 -> rows +8..15 (C layout)
            int mbase = mtile * 256 + wave * 32 + sub * 16 + lh * 8;
#pragma unroll
            for (int r = 0; r < 8; ++r) {
                int mr = mbase + r;
                int bo = mr >> 8;
                int to = mr & (TT - 1);
                idxw[((size_t)bo * CC + c) * TT + to] = bidx[r];
            }
        }
    }
}

// ---------------------------------------------------------------------------
// Kernel 3: gather values, mean over codebooks, softmax over T.
// One block per (v, b) column; t = threadIdx.x.
// ---------------------------------------------------------------------------
__global__ __launch_bounds__(256)
void gather_softmax_kernel(const float* __restrict__ values, const int* __restrict__ idxw,
                           float* __restrict__ out) {
    const int v = blockIdx.x;       // 0..63
    const int b = blockIdx.y;       // 0..15
    const int t = threadIdx.x;      // 0..255

    float acc = 0.f;
#pragma unroll
    for (int c = 0; c < CC; ++c) {
        int id = idxw[((size_t)b * CC + c) * TT + t];
        acc += values[((size_t)c * KK + id) * VV + v];
    }
    acc *= 0.125f;                  // mean over CC=8

    __shared__ float red[8];
    const int lane = t & 31, wv = t >> 5;

    // block max
    float mx = acc;
    for (int off = 16; off >= 1; off >>= 1) mx = fmaxf(mx, __shfl_xor(mx, off, 32));
    if (lane == 0) red[wv] = mx;
    __syncthreads();
    mx = red[0];
#pragma unroll
    for (int w = 1; w < 8; ++w) mx = fmaxf(mx, red[w]);
    __syncthreads();

    // block sum of exp
    float e = __expf(acc - mx);
    float s = e;
    for (int off = 16; off >= 1; off >>= 1) s += __shfl_xor(s, off, 32);
    if (lane == 0) red[wv] = s;
    __syncthreads();
    s = 0.f;
#pragma unroll
    for (int w = 0; w < 8; ++w) s += red[w];

    out[((size_t)b * TT + t) * VV + v] = e / s;
}

// ---------------------------------------------------------------------------
extern "C" void kernel_launch(void* const* d_in, const int* in_sizes, int n_in,
                              void* d_out, int out_size, void* d_ws, size_t ws_size,
                              hipStream_t stream) {
    const float* batch  = (const float*)d_in[0];   // [B,C,T,D]
    const float* keys   = (const float*)d_in[1];   // [C,K,D]
    const float* values = (const float*)d_in[2];   // [C,K,V]
    float* out = (float*)d_out;                    // [B,T,V]

    float* k2w  = (float*)d_ws;                                // C*K floats
    int*   idxw = (int*)((char*)d_ws + (size_t)CC * KK * sizeof(float)); // B*C*T ints

    k2_kernel<<<(CC * KK + 255) / 256, 256, 0, stream>>>(keys, k2w);
    argmax_kernel<<<dim3(16, CC), 256, 0, stream>>>(batch, keys, k2w, idxw);
    gather_softmax_kernel<<<dim3(VV, BB), 256, 0, stream>>>(values, idxw, out);
}